// ETKFWeightsModule_40518721471146
// MI455X (gfx1250) — compile-verified
//
#include <hip/hip_runtime.h>
#include <hip/hip_bf16.h>

typedef float v2f __attribute__((ext_vector_type(2)));
typedef float v8f __attribute__((ext_vector_type(8)));

#define ENS 64
#define LDA 66   // LDS row stride (floats); 64+2 keeps A(float2) and B(column) loads bank-conflict-free

// Cooperative 64x64x64 fp32 GEMM on WMMA: C = scale * A * op(B) + diag * I
// A,B,C are 64xLDA row-major LDS buffers. transB=true -> op(B)=B^T (contiguous frag loads).
// Called by all 256 threads (8 waves); wave w computes tiles 2w and 2w+1 of the 4x4 tile grid.
__device__ __forceinline__ void gemm64(const float* A, const float* B, float* C,
                                       bool transB, float scale, float diag, int tid) {
  const int lane = tid & 31;
  const int wave = tid >> 5;
  const int lr   = lane & 15;   // row (A/M) or col (B/N, C/N) within tile
  const int half = lane >> 4;   // lane half selects K-pair / M-offset
  #pragma unroll
  for (int t = 0; t < 2; ++t) {
    const int tile = wave * 2 + t;
    const int mt = (tile >> 2) << 4;
    const int nt = (tile & 3) << 4;
    v8f acc = {};
    #pragma unroll
    for (int kb = 0; kb < 64; kb += 4) {
      const int k2 = kb + 2 * half;
      // A 16x4 frag: element (m,k) at lane = m + 16*(k>>1)... -> lane lr holds row mt+lr,
      // vgpr = k&1, half picks k pair: contiguous float2 A[mt+lr][k2..k2+1]
      v2f a = *(const v2f*)(A + (mt + lr) * LDA + k2);
      v2f b;
      if (transB) {
        // op(B)[k][n] = B[n][k] -> contiguous float2 per lane
        b = *(const v2f*)(B + (nt + lr) * LDA + k2);
      } else {
        // B 4x16 frag: row K = vgpr + 2*half, striped across lanes (N = lr)
        b.x = B[(k2 + 0) * LDA + nt + lr];
        b.y = B[(k2 + 1) * LDA + nt + lr];
      }
      acc = __builtin_amdgcn_wmma_f32_16x16x4_f32(false, a, false, b,
                                                  (short)0, acc, false, false);
    }
    // C/D layout: VGPR v holds M = mt + v + 8*half, N = nt + lr
    #pragma unroll
    for (int v = 0; v < 8; ++v) {
      const int m = mt + v + 8 * half;
      const int n = nt + lr;
      float r = scale * acc[v];
      if (m == n) r += diag;
      C[m * LDA + n] = r;
    }
  }
}

__global__ __launch_bounds__(256) void etkf_ns_kernel(
    const float* __restrict__ perts, const float* __restrict__ obs,
    const float* __restrict__ inf_factor,
    float* __restrict__ out_weights, float* __restrict__ out_wmean,
    float* __restrict__ out_wperts, float* __restrict__ out_cov) {
  __shared__ float sB0[ENS * LDA];
  __shared__ float sB1[ENS * LDA];
  __shared__ float sB2[ENS * LDA];
  __shared__ float sB3[ENS * LDA];
  __shared__ float sObs[ENS];
  __shared__ float sKobs[ENS];
  __shared__ float sRed[ENS];
  __shared__ float sC;

  const int tid = threadIdx.x;
  const int b   = blockIdx.x;
  const float* P = perts + (size_t)b * ENS * ENS;

  // ---- load P (16KB) and obs into LDS, coalesced ----
  #pragma unroll
  for (int i = 0; i < 16; ++i) {
    int idx = tid + i * 256;
    sB0[(idx >> 6) * LDA + (idx & 63)] = P[idx];
  }
  if (tid < ENS) sObs[tid] = obs[(size_t)b * ENS + tid];
  __syncthreads();

  // ---- kernel_obs = P @ obs^T (64 dot products; overlaps with GEMM below) ----
  if (tid < ENS) {
    float s = 0.f;
    #pragma unroll 8
    for (int j = 0; j < ENS; ++j) s += sB0[tid * LDA + j] * sObs[j];
    sKobs[tid] = s;
  }

  const float reg = (float)(ENS - 1) / inf_factor[0];

  // ---- S = P P^T + reg*I  (SPD, replaces eigh regularization) -> sB3 ----
  gemm64(sB0, sB0, sB3, /*transB=*/true, 1.0f, reg, tid);
  __syncthreads();

  // ---- c = ||S||_inf >= lambda_max : scaling so A = S/c has spectrum in (0,1] ----
  if (tid < ENS) {
    float s = 0.f;
    for (int j = 0; j < ENS; ++j) s += fabsf(sB3[tid * LDA + j]);
    sRed[tid] = s;
  }
  __syncthreads();
  if (tid == 0) {
    float m = 0.f;
    for (int i = 0; i < ENS; ++i) m = fmaxf(m, sRed[i]);
    sC = m;
  }
  __syncthreads();
  const float c    = sC;
  const float invc = 1.0f / c;

  // ---- Y0 = S/c -> sB1, Z0 = I -> sB2 ----
  #pragma unroll
  for (int i = 0; i < 16; ++i) {
    int idx = tid + i * 256;
    int r = idx >> 6, cl = idx & 63;
    sB1[r * LDA + cl] = sB3[r * LDA + cl] * invc;
    sB2[r * LDA + cl] = (r == cl) ? 1.0f : 0.0f;
  }
  __syncthreads();

  // ---- coupled Newton-Schulz: Y->A^{1/2}, Z->A^{-1/2} (all iterates commute/symmetric) ----
  float* pY = sB1;
  float* pZ = sB2;
  float* pT = sB3;
  float* pW = sB0;  // P is dead now
  #pragma unroll 1
  for (int it = 0; it < 12; ++it) {
    gemm64(pZ, pY, pT, false, -1.0f, 3.0f, tid);  // T = 3I - Z*Y
    __syncthreads();
    gemm64(pY, pT, pW, false, 0.5f, 0.0f, tid);   // Ynew = 0.5*Y*T
    __syncthreads();
    gemm64(pZ, pT, pY, false, 0.5f, 0.0f, tid);   // Znew = 0.5*Z*T (into dead Y buffer)
    __syncthreads();
    float* nY = pW; float* nZ = pY; float* nW = pZ;
    pY = nY; pZ = nZ; pW = nW;                    // pT reused every iteration
  }

  // ---- cov = S^{-1} = Z*Z^T / c -> pT ----
  gemm64(pZ, pZ, pT, /*transB=*/true, invc, 0.0f, tid);
  __syncthreads();

  // ---- w_mean = cov @ kernel_obs ----
  if (tid < ENS) {
    float s = 0.f;
    #pragma unroll 8
    for (int j = 0; j < ENS; ++j) s += pT[tid * LDA + j] * sKobs[j];
    sRed[tid] = s;
  }
  __syncthreads();

  // ---- outputs: w_perts = sqrt(63/c)*Z ; weights = w_perts + w_mean[row] ----
  const float spf  = sqrtf((float)(ENS - 1) * invc);
  const size_t base = (size_t)b * ENS * ENS;
  #pragma unroll
  for (int i = 0; i < 16; ++i) {
    int idx = tid + i * 256;
    int r = idx >> 6, cl = idx & 63;
    float wp = spf * pZ[r * LDA + cl];
    float cv = pT[r * LDA + cl];
    out_wperts[base + idx]  = wp;
    out_cov[base + idx]     = cv;
    out_weights[base + idx] = wp + sRed[r];
  }
  if (tid < ENS) out_wmean[(size_t)b * ENS + tid] = sRed[tid];
}

extern "C" void kernel_launch(void* const* d_in, const int* in_sizes, int n_in,
                              void* d_out, int out_size, void* d_ws, size_t ws_size,
                              hipStream_t stream) {
  const float* perts = (const float*)d_in[0];  // [B,64,64]
  const float* obs   = (const float*)d_in[1];  // [B,1,64]
  const float* inf   = (const float*)d_in[2];  // scalar
  const int B = in_sizes[0] / (ENS * ENS);

  float* out       = (float*)d_out;
  float* o_weights = out;                                   // [B,64,64]
  float* o_wmean   = o_weights + (size_t)B * ENS * ENS;     // [B,64,1]
  float* o_wperts  = o_wmean   + (size_t)B * ENS;           // [B,64,64]
  float* o_cov     = o_wperts  + (size_t)B * ENS * ENS;     // [B,64,64]

  etkf_ns_kernel<<<B, 256, 0, stream>>>(perts, obs, inf,
                                        o_weights, o_wmean, o_wperts, o_cov);
}